// Output_5248450035879
// MI455X (gfx1250) — compile-verified
//
#include <hip/hip_runtime.h>

typedef float v2f __attribute__((ext_vector_type(2)));
typedef float v8f __attribute__((ext_vector_type(8)));

// ---------------------------------------------------------------------------
// Kernel 1: per-block masked-node counts (block = 256 threads over 256 nodes)
// ---------------------------------------------------------------------------
__global__ void k_count(const int* __restrict__ symbols,
                        const int* __restrict__ symbuf, int T, int N,
                        int* __restrict__ blockCounts) {
    __shared__ int cnt;
    const int tid = threadIdx.x;
    if (tid == 0) cnt = 0;
    __syncthreads();
    const int i = blockIdx.x * blockDim.x + tid;
    int m = 0;
    if (i < N) {
        const int s = symbols[i];
        for (int t = 0; t < T; ++t) m |= (s == symbuf[t]);
    }
    const unsigned bal = __builtin_amdgcn_ballot_w32(m != 0);
    if ((tid & 31) == 0) atomicAdd(&cnt, __popc(bal));
    __syncthreads();
    if (tid == 0) blockCounts[blockIdx.x] = cnt;
}

// ---------------------------------------------------------------------------
// Kernel 2a: one-block Hillis-Steele exclusive scan (nb <= 1024)
// ---------------------------------------------------------------------------
__global__ void k_scan_par(const int* __restrict__ cnt, int* __restrict__ off,
                           int nb) {
    __shared__ int sm[1024];
    const int tid = threadIdx.x;
    const int v = (tid < nb) ? cnt[tid] : 0;
    sm[tid] = v;
    __syncthreads();
    for (int d = 1; d < 1024; d <<= 1) {
        const int t = (tid >= d) ? sm[tid - d] : 0;
        __syncthreads();
        sm[tid] += t;
        __syncthreads();
    }
    if (tid < nb) off[tid] = sm[tid] - v;  // exclusive
}

// Kernel 2b: serial fallback for nb > 1024 (not hit for N=200k)
__global__ void k_scan_ser(const int* __restrict__ cnt, int* __restrict__ off,
                           int nb) {
    if (threadIdx.x == 0 && blockIdx.x == 0) {
        int run = 0;
        for (int b = 0; b < nb; ++b) { off[b] = run; run += cnt[b]; }
    }
}

// ---------------------------------------------------------------------------
// Kernel 3: ordered stream compaction -> idx[K], mid[K] = meta_node_id[idx]
// ---------------------------------------------------------------------------
__global__ void k_compact(const int* __restrict__ symbols,
                          const int* __restrict__ symbuf, int T, int N,
                          const int* __restrict__ meta,
                          const int* __restrict__ blockOff, int K,
                          int* __restrict__ idxOut, int* __restrict__ midOut) {
    __shared__ int waveOff[8];
    const int tid = threadIdx.x;
    const int i = blockIdx.x * blockDim.x + tid;
    int m = 0;
    if (i < N) {
        const int s = symbols[i];
        for (int t = 0; t < T; ++t) m |= (s == symbuf[t]);
    }
    const int lane = tid & 31;
    const int wid = tid >> 5;
    const unsigned bal = __builtin_amdgcn_ballot_w32(m != 0);
    const int waveCnt = __popc(bal);
    const int rank = __popc(bal & ((lane == 31) ? 0x7fffffffu
                                                : ((1u << lane) - 1u)));
    if (lane == 0) waveOff[wid] = waveCnt;
    __syncthreads();
    if (tid == 0) {
        int run = 0;
        const int nw = (int)(blockDim.x >> 5);
        for (int w = 0; w < nw; ++w) { int t = waveOff[w]; waveOff[w] = run; run += t; }
    }
    __syncthreads();
    if (m) {
        const int pos = blockOff[blockIdx.x] + waveOff[wid] + rank;
        if (pos < K) {
            idxOut[pos] = i;
            midOut[pos] = meta[i];
        }
    }
}

// ---------------------------------------------------------------------------
// Kernel 4: main WMMA kernel. One wave (32 threads) per 16-row tile.
// dot_jl[m] = diag(A_jl . W_j^T) computed with V_WMMA_F32_16X16X4_F32.
// f32 A 16x4 layout: lanes 0-15 = rows M=0..15 @ K={0,1}; lanes 16-31 same
// rows @ K={2,3}.  f32 B 4x16 layout mirrors it with lane = column N.
// => every lane loads one float2 per operand per K-step.
// ---------------------------------------------------------------------------
__global__ __launch_bounds__(32) void k_main(
    const float* __restrict__ f00, const float* __restrict__ f01,
    const float* __restrict__ f10, const float* __restrict__ f11,
    const float* __restrict__ f20, const float* __restrict__ f21,
    const float* __restrict__ w0, const float* __restrict__ w1,
    const float* __restrict__ w2, const float* __restrict__ b0,
    const float* __restrict__ b1, const float* __restrict__ b2,
    const int* __restrict__ idx, const int* __restrict__ mid,
    int D, int K, float* __restrict__ out) {
    __shared__ float smem[16 * 16];

    const int lane = threadIdx.x;      // 0..31
    const int hi   = lane >> 4;        // 0: K-offsets {0,1}; 1: K-offsets {2,3}
    const int m    = lane & 15;        // tile row (A) == tile column (B)

    const int r  = blockIdx.x * 16 + m;
    const int rc = (r < K) ? r : (K - 1);   // clamp: keeps EXEC all-ones
    const int irow = idx[rc];
    const int mrow = mid[rc];

    const size_t fo = (size_t)irow * (size_t)D + (size_t)(hi * 2);
    const size_t wo = (size_t)mrow * (size_t)D + (size_t)(hi * 2);
    const float* A0 = f00 + fo;  const float* A1 = f01 + fo;
    const float* A2 = f10 + fo;  const float* A3 = f11 + fo;
    const float* A4 = f20 + fo;  const float* A5 = f21 + fo;
    const float* B0 = w0 + wo;   const float* B1 = w1 + wo;
    const float* B2 = w2 + wo;

    v8f c00 = {}, c01 = {}, c10 = {}, c11 = {}, c20 = {}, c21 = {};

    const int steps = D >> 2;  // K advances 4 per WMMA
#pragma unroll 8
    for (int kb = 0; kb < steps; ++kb) {
        const int off = kb * 4;
        // weights: hot in L2 (3 MB tables, ~50x reuse) -> regular temporal
        const v2f bv0 = *(const v2f*)(B0 + off);
        const v2f bv1 = *(const v2f*)(B1 + off);
        const v2f bv2 = *(const v2f*)(B2 + off);
        // features: streamed once (300 MB) -> non-temporal, keep L2 for weights
        const v2f a0 = __builtin_nontemporal_load((const v2f*)(A0 + off));
        const v2f a1 = __builtin_nontemporal_load((const v2f*)(A1 + off));
        const v2f a2 = __builtin_nontemporal_load((const v2f*)(A2 + off));
        const v2f a3 = __builtin_nontemporal_load((const v2f*)(A3 + off));
        const v2f a4 = __builtin_nontemporal_load((const v2f*)(A4 + off));
        const v2f a5 = __builtin_nontemporal_load((const v2f*)(A5 + off));

        c00 = __builtin_amdgcn_wmma_f32_16x16x4_f32(false, a0, false, bv0,
                                                    (short)0, c00, false, false);
        c01 = __builtin_amdgcn_wmma_f32_16x16x4_f32(false, a1, false, bv0,
                                                    (short)0, c01, false, false);
        c10 = __builtin_amdgcn_wmma_f32_16x16x4_f32(false, a2, false, bv1,
                                                    (short)0, c10, false, false);
        c11 = __builtin_amdgcn_wmma_f32_16x16x4_f32(false, a3, false, bv1,
                                                    (short)0, c11, false, false);
        c20 = __builtin_amdgcn_wmma_f32_16x16x4_f32(false, a4, false, bv2,
                                                    (short)0, c20, false, false);
        c21 = __builtin_amdgcn_wmma_f32_16x16x4_f32(false, a5, false, bv2,
                                                    (short)0, c21, false, false);
    }

    // Sum the six 16x16 accumulators, spill to LDS, pull the diagonal.
    const v8f s = c00 + c01 + c10 + c11 + c20 + c21;
#pragma unroll
    for (int v = 0; v < 8; ++v) {
        const int row = hi * 8 + v;        // C/D layout: VGPR v -> M = v | 8+v
        smem[row * 16 + m] = s[v];
    }
    __syncthreads();

    if (lane < 16 && r < K) {
        const float diag = smem[lane * 16 + lane];
        const float bs = b0[mrow] + b1[mrow] + b2[mrow];
        out[r] = (diag + 2.0f * bs) * (1.0f / 6.0f);
    }
}

// ---------------------------------------------------------------------------
extern "C" void kernel_launch(void* const* d_in, const int* in_sizes, int n_in,
                              void* d_out, int out_size, void* d_ws,
                              size_t ws_size, hipStream_t stream) {
    const float* f00 = (const float*)d_in[0];
    const float* f01 = (const float*)d_in[1];
    const float* f10 = (const float*)d_in[2];
    const float* f11 = (const float*)d_in[3];
    const float* f20 = (const float*)d_in[4];
    const float* f21 = (const float*)d_in[5];
    const float* w0  = (const float*)d_in[6];
    const float* w1  = (const float*)d_in[7];
    const float* w2  = (const float*)d_in[8];
    const float* b0  = (const float*)d_in[9];
    const float* b1  = (const float*)d_in[10];
    const float* b2  = (const float*)d_in[11];
    const int* symbols = (const int*)d_in[12];
    const int* meta    = (const int*)d_in[13];
    const int* symbuf  = (const int*)d_in[14];
    // d_in[15] (num_masked) mirrored by out_size; not needed on device.

    const int M = in_sizes[9];            // rows of b0 -> 1000
    const int D = in_sizes[6] / M;        // 256
    const int N = in_sizes[0] / D;        // 200000
    const int T = in_sizes[14];           // 2
    const int K = out_size;               // 50000
    if (K <= 0) return;

    const int nb = (N + 255) / 256;

    int* ws     = (int*)d_ws;
    int* idxBuf = ws;                 // [K]
    int* midBuf = ws + K;             // [K]
    int* bCnt   = ws + 2 * K;         // [nb]
    int* bOff   = ws + 2 * K + nb;    // [nb]

    k_count<<<nb, 256, 0, stream>>>(symbols, symbuf, T, N, bCnt);
    if (nb <= 1024)
        k_scan_par<<<1, 1024, 0, stream>>>(bCnt, bOff, nb);
    else
        k_scan_ser<<<1, 1, 0, stream>>>(bCnt, bOff, nb);
    k_compact<<<nb, 256, 0, stream>>>(symbols, symbuf, T, N, meta, bOff, K,
                                      idxBuf, midBuf);

    const int tiles = (K + 15) / 16;
    k_main<<<tiles, 32, 0, stream>>>(f00, f01, f10, f11, f20, f21, w0, w1, w2,
                                     b0, b1, b2, idxBuf, midBuf, D, K,
                                     (float*)d_out);
}